// ChamferLoss_61727269978572
// MI455X (gfx1250) — compile-verified
//
#include <hip/hip_runtime.h>

typedef __attribute__((ext_vector_type(2))) float v2f;
typedef __attribute__((ext_vector_type(8))) float v8f;

#define POS_W   1.0f
#define ROT_W   0.5f
#define SCALE_W 0.5f
#define OPAC_W  0.3f
#define SH_W    0.2f

__device__ __forceinline__ float clamp_pos(float a) { return a < 0.0f ? 0.0f : a; }

// ---------------------------------------------------------------------------
// Pass 1/2: for each "own" point, min (and optionally argmin) of squared
// distance to all "other" points, via V_WMMA_F32_16X16X4_F32.
// One wave owns a 16-row tile of "own"; loops over all 16-col tiles of "other".
// A fragment pre-scaled by -2 and C preloaded with |a|^2+|b|^2, so the WMMA
// emits squared distances directly: D = (-2A)B + (an+bn) = |a-b|^2.
// ---------------------------------------------------------------------------
template <bool WITH_IDX>
__global__ __launch_bounds__(256) void chamfer_min_kernel(
    const float* __restrict__ own_xyz,    // [B, ownCount, 3]
    const float* __restrict__ other_xyz,  // [B, otherCount, 3]
    int ownCount, int otherCount, int nBatch,
    float* __restrict__ out_min,          // [B, ownCount]  (sqrt'ed at the end)
    int* __restrict__ out_idx)            // [B, ownCount]  (only if WITH_IDX)
{
  const int lane = threadIdx.x & 31;
  const int hi   = lane >> 4;   // 0: lanes 0-15, 1: lanes 16-31
  const int l16  = lane & 15;

  const int tilesPerBatch = ownCount >> 4;
  const int wave = blockIdx.x * (blockDim.x >> 5) + (threadIdx.x >> 5);
  if (wave >= tilesPerBatch * nBatch) return;    // uniform per-wave exit
  const int b  = wave / tilesPerBatch;
  const int m0 = (wave - b * tilesPerBatch) << 4;

  const float* own = own_xyz   + (size_t)b * ownCount   * 3;
  const float* oth = other_xyz + (size_t)b * otherCount * 3;

  // A fragment: 16x4 f32 (rows m0..m0+15, K = {x,y,z,0}), pre-scaled by -2.
  // ISA layout: lanes 0-15 -> v0=K0, v1=K1 ; lanes 16-31 -> v0=K2, v1=K3.
  v2f afrag;
  {
    const float* p = own + (size_t)(m0 + l16) * 3;
    float x = p[0], y = p[1], z = p[2];
    afrag.x = -2.0f * (hi ? z : x);
    afrag.y = -2.0f * (hi ? 0.0f : y);
  }

  // |own|^2 for the 8 rows this lane's D registers map to (M = m0 + r + 8*hi).
  float an[8];
#pragma unroll
  for (int r = 0; r < 8; ++r) {
    const float* q = own + (size_t)(m0 + r + 8 * hi) * 3;
    an[r] = q[0] * q[0] + q[1] * q[1] + q[2] * q[2];
  }

  float mval[8];
  int   midx[8];
#pragma unroll
  for (int r = 0; r < 8; ++r) { mval[r] = 3.402823466e+38f; midx[r] = 0; }

  const int nTiles = otherCount >> 4;
  for (int t = 0; t < nTiles; ++t) {
    const int j0 = t << 4;
    const float* q = oth + (size_t)(j0 + l16) * 3;
    float x = q[0], y = q[1], z = q[2];
    float bn = x * x + y * y + z * z;      // |other(n_lane)|^2
    // B fragment: 4x16 f32, symmetric layout to A.
    v2f bfrag;
    bfrag.x = hi ? z : x;
    bfrag.y = hi ? 0.0f : y;
    // C preloaded with |a|^2 + |b|^2 so the WMMA yields |a-b|^2 directly.
    v8f c;
#pragma unroll
    for (int r = 0; r < 8; ++r) c[r] = an[r] + bn;
    v8f d2 = __builtin_amdgcn_wmma_f32_16x16x4_f32(
        false, afrag, false, bfrag, (short)0, c, false, false);
    const int idx = j0 + l16;
#pragma unroll
    for (int r = 0; r < 8; ++r) {
      if (WITH_IDX) {
        bool better = d2[r] < mval[r];     // strict < keeps earliest n
        mval[r] = better ? d2[r] : mval[r];
        midx[r] = better ? idx : midx[r];
      } else {
        mval[r] = fminf(mval[r], d2[r]);   // v_min_num_f32, no cndmask
      }
    }
  }

  // Reduce across the 16 lanes sharing each output row; xor masks 1..8 stay
  // within each half-wave (matches D-matrix lane split).
#pragma unroll
  for (int off = 1; off <= 8; off <<= 1) {
#pragma unroll
    for (int r = 0; r < 8; ++r) {
      float ov = __shfl_xor(mval[r], off, 32);
      if (WITH_IDX) {
        int oi = __shfl_xor(midx[r], off, 32);
        bool take = (ov < mval[r]) || ((ov == mval[r]) && (oi < midx[r]));
        mval[r] = take ? ov : mval[r];
        midx[r] = take ? oi : midx[r];
      } else {
        mval[r] = fminf(mval[r], ov);
      }
    }
  }

  if (l16 == 0) {
#pragma unroll
    for (int r = 0; r < 8; ++r) {
      int m = m0 + r + 8 * hi;
      size_t o = (size_t)b * ownCount + m;
      out_min[o] = sqrtf(clamp_pos(mval[r]));
      if (WITH_IDX) out_idx[o] = midx[r];
    }
  }
}

// ---------------------------------------------------------------------------
// Pass 3: gather matched attributes, fold all normalizations into one scalar
// per thread, LDS block-reduce -> per-block partial (deterministic).
// ---------------------------------------------------------------------------
__global__ __launch_bounds__(256) void chamfer_loss_reduce(
    const float* __restrict__ min_out, const int* __restrict__ match_idx,
    const float* __restrict__ min_in,
    const float* __restrict__ in_rot, const float* __restrict__ in_scale,
    const float* __restrict__ in_opac, const float* __restrict__ in_dc,
    const float* __restrict__ in_rest,
    const float* __restrict__ out_rot, const float* __restrict__ out_scale,
    const float* __restrict__ out_opac, const float* __restrict__ out_dc,
    const float* __restrict__ out_rest,
    float* __restrict__ partials, int Bn, int Nn, int Mn)
{
  const int i  = blockIdx.x * blockDim.x + threadIdx.x;
  const int BM = Bn * Mn, BN = Bn * Nn;
  const float invBM = 1.0f / (float)BM;
  const float invBN = 1.0f / (float)BN;

  float acc = 0.0f;
  if (i < BM) {
    int b = i / Mn;
    size_t j = (size_t)b * Nn + match_idx[i];

    // pos (out->in half)
    acc += POS_W * 0.5f * invBM * min_out[i];
    // rot: 1 - |dot4|
    float dot = 0.0f;
#pragma unroll
    for (int k = 0; k < 4; ++k) dot += out_rot[(size_t)i * 4 + k] * in_rot[j * 4 + k];
    acc += ROT_W * invBM * (1.0f - fabsf(dot));
    // scale L1 (mean over 3 comps)
    float s = 0.0f;
#pragma unroll
    for (int k = 0; k < 3; ++k) s += fabsf(out_scale[(size_t)i * 3 + k] - in_scale[j * 3 + k]);
    acc += SCALE_W * (invBM / 3.0f) * s;
    // opacity
    acc += OPAC_W * invBM * fabsf(out_opac[i] - in_opac[j]);
    // SH
    float dc = 0.0f;
#pragma unroll
    for (int k = 0; k < 3; ++k) dc += fabsf(out_dc[(size_t)i * 3 + k] - in_dc[j * 3 + k]);
    float rest = 0.0f;
    for (int k = 0; k < 45; ++k) rest += fabsf(out_rest[(size_t)i * 45 + k] - in_rest[j * 45 + k]);
    acc += SH_W * ((invBM / 3.0f) * dc + (invBM / 45.0f) * rest);
  }
  if (i < BN) acc += POS_W * 0.5f * invBN * min_in[i];  // pos (in->out half)

  __shared__ float red[256];
  red[threadIdx.x] = acc;
  __syncthreads();
  for (int s2 = 128; s2 > 0; s2 >>= 1) {
    if (threadIdx.x < s2) red[threadIdx.x] += red[threadIdx.x + s2];
    __syncthreads();
  }
  if (threadIdx.x == 0) partials[blockIdx.x] = red[0];
}

__global__ void chamfer_finalize(const float* __restrict__ partials, int n,
                                 float* __restrict__ out)
{
  if (threadIdx.x == 0 && blockIdx.x == 0) {
    float s = 0.0f;
    for (int k = 0; k < n; ++k) s += partials[k];  // fixed order: deterministic
    out[0] = s;
  }
}

// ---------------------------------------------------------------------------
extern "C" void kernel_launch(void* const* d_in, const int* in_sizes, int n_in,
                              void* d_out, int out_size, void* d_ws, size_t ws_size,
                              hipStream_t stream)
{
  (void)n_in; (void)out_size; (void)ws_size;

  const float* in_xyz    = (const float*)d_in[0];
  const float* in_rot    = (const float*)d_in[1];
  const float* in_scale  = (const float*)d_in[2];
  const float* in_opac   = (const float*)d_in[3];
  const float* in_dc     = (const float*)d_in[4];
  const float* in_rest   = (const float*)d_in[5];
  const float* out_xyz   = (const float*)d_in[6];
  const float* out_rot   = (const float*)d_in[7];
  const float* out_scale = (const float*)d_in[8];
  const float* out_opac  = (const float*)d_in[9];
  const float* out_dc    = (const float*)d_in[10];
  const float* out_rest  = (const float*)d_in[11];

  const int Bn = 2;
  const int Nn = in_sizes[0] / (Bn * 3);   // 8192
  const int Mn = in_sizes[6] / (Bn * 3);   // 8192
  const int BM = Bn * Mn, BN = Bn * Nn;

  // workspace layout
  float* min_out  = (float*)d_ws;                 // BM floats
  int*   midx     = (int*)(min_out + BM);         // BM ints
  float* min_in   = (float*)(midx + BM);          // BN floats
  float* partials = min_in + BN;                  // <= 64 floats

  {
    int waves  = Bn * (Mn >> 4);
    int blocks = (waves + 7) / 8;                 // 8 waves / 256-thread block
    chamfer_min_kernel<true><<<blocks, 256, 0, stream>>>(
        out_xyz, in_xyz, Mn, Nn, Bn, min_out, midx);
  }
  {
    int waves  = Bn * (Nn >> 4);
    int blocks = (waves + 7) / 8;
    chamfer_min_kernel<false><<<blocks, 256, 0, stream>>>(
        in_xyz, out_xyz, Nn, Mn, Bn, min_in, nullptr);
  }
  {
    int total  = BM > BN ? BM : BN;
    int blocks = (total + 255) / 256;
    chamfer_loss_reduce<<<blocks, 256, 0, stream>>>(
        min_out, midx, min_in,
        in_rot, in_scale, in_opac, in_dc, in_rest,
        out_rot, out_scale, out_opac, out_dc, out_rest,
        partials, Bn, Nn, Mn);
    chamfer_finalize<<<1, 1, 0, stream>>>(partials, blocks, (float*)d_out);
  }
}